// RoutingLayer_4346506904218
// MI455X (gfx1250) — compile-verified
//
#include <hip/hip_runtime.h>
#include <hip/hip_bf16.h>
#include <stdint.h>

// RoutingLayer on MI455X (gfx1250):
//  - TDM gather-mode loads pull 32 neighbor rows (512B each) into LDS (2 descriptors,
//    16-bit row indices), tracked with TENSORcnt.
//  - All 6 routing iterations run out of LDS; u1/u2 einsums are done with
//    v_wmma_f32_16x16x4_f32 as U12[128,16] = Z^T[128,32] x [p1|p2][32,16],
//    extracting the block diagonal.
//  - One wave per node; wave32 shuffle reductions for softmax-over-neighbors.

#define DIM   128
#define KCAPS 8
#define DD    16
#define MNBR  32
#define WPB   2   // waves per block

typedef float v2f __attribute__((ext_vector_type(2)));
typedef float v4f __attribute__((ext_vector_type(4)));
typedef float v8f __attribute__((ext_vector_type(8)));
typedef unsigned int u32x4 __attribute__((ext_vector_type(4)));
typedef int i32x4 __attribute__((ext_vector_type(4)));
typedef int i32x8 __attribute__((ext_vector_type(8)));

struct alignas(128) WaveMem {
  float z[MNBR * DIM];   // 16KB: gathered + per-capsule normalized neighbor rows
  float ub[DIM];         // current u
  float ubf[DIM];        // u_before (normalized own features)
  float ut1[DIM];        // u1 staging (from WMMA diagonal)
  float ut2[DIM];        // u2 staging
  float pb[MNBR * 16];   // per-neighbor [p1(8) | p2(8)]
};

__device__ inline float dot4(v4f a, v4f b) {
  return a.x * b.x + a.y * b.y + a.z * b.z + a.w * b.w;
}

__global__ __launch_bounds__(WPB * 32)
void routing_caps_kernel(const float* __restrict__ x,
                         const int* __restrict__ nbr,
                         const float* __restrict__ paramp,
                         const int* __restrict__ iterp,
                         float* __restrict__ out,
                         int n) {
  __shared__ WaveMem smem[WPB];
  const int lane = threadIdx.x & 31;
  const int wv = __builtin_amdgcn_readfirstlane((int)(threadIdx.x >> 5));
  int node = (int)blockIdx.x * WPB + wv;
  const bool valid = (node < n);
  if (!valid) node = 0;  // redundant work, keeps EXEC full for WMMA/barriers
  WaveMem* sm = &smem[wv];

  // ---------------- TDM gather: 32 neighbor rows -> LDS ----------------
  const unsigned* nbp = (const unsigned*)(nbr + (size_t)node * MNBR);
  unsigned idxw[16];
#pragma unroll
  for (int i = 0; i < 16; ++i) {
    unsigned pk = (nbp[2 * i] & 0xFFFFu) | (nbp[2 * i + 1] << 16);
    idxw[i] = (unsigned)__builtin_amdgcn_readfirstlane((int)pk);
  }

  const uint64_t gaddr = (uint64_t)(uintptr_t)x;
  const unsigned lds0 = (unsigned)(uintptr_t)(&sm->z[0]);

  // D# group1: data_size=4B, tensor_dim0=128, tensor_dim1=n, tile_dim0=128,
  // tile_dim1=16 valid indices, tensor_dim0_stride=128 elements.
  i32x8 g1;
  g1[0] = 0x00020000;                 // workgroup_mask=0, data_size=2 (4B)
  g1[1] = (int)(128u << 16);          // tensor_dim0[15:0] at [31:16]
  g1[2] = (int)(((unsigned)n & 0xFFFFu) << 16);  // tensor_dim1[15:0] at [31:16]
  g1[3] = (int)(128u << 16);          // tile_dim0 at [31:16]
  g1[4] = 16;                         // tile_dim1 = #valid indices
  g1[5] = 128;                        // tensor_dim0_stride[31:0]
  g1[6] = 0;
  g1[7] = 0;

  u32x4 g0;
  g0[0] = 1u | (1u << 31);            // count=1, gather_mode=1, 16-bit indices
  g0[1] = lds0;                       // LDS dest (rows 0..15)
  g0[2] = (unsigned)(gaddr & 0xFFFFFFFFu);
  g0[3] = (unsigned)((gaddr >> 32) & 0x01FFFFFFu) | 0x80000000u;  // type=2

  const i32x8 gz = {0, 0, 0, 0, 0, 0, 0, 0};  // extra operand on this toolchain

  i32x4 g2 = {(int)idxw[0], (int)idxw[1], (int)idxw[2], (int)idxw[3]};
  i32x4 g3 = {(int)idxw[4], (int)idxw[5], (int)idxw[6], (int)idxw[7]};
  __builtin_amdgcn_tensor_load_to_lds(g0, g1, g2, g3, gz, 0);

  g0[1] = lds0 + 16u * 512u;          // LDS dest (rows 16..31)
  i32x4 h2 = {(int)idxw[8], (int)idxw[9], (int)idxw[10], (int)idxw[11]};
  i32x4 h3 = {(int)idxw[12], (int)idxw[13], (int)idxw[14], (int)idxw[15]};
  __builtin_amdgcn_tensor_load_to_lds(g0, g1, h2, h3, gz, 0);

  __builtin_amdgcn_s_wait_tensorcnt(0);
  __syncthreads();

  // ---------------- per-capsule L2-normalize gathered rows ----------------
  float* zr = sm->z + lane * DIM;   // lane <-> neighbor m
#pragma unroll
  for (int c = 0; c < KCAPS; ++c) {
    v4f* zc = (v4f*)(zr + c * DD);
    v4f a0 = zc[0], a1 = zc[1], a2 = zc[2], a3 = zc[3];
    float ss = dot4(a0, a0) + dot4(a1, a1) + dot4(a2, a2) + dot4(a3, a3);
    float inv = 1.0f / fmaxf(sqrtf(ss), 1e-12f);
    zc[0] = a0 * inv; zc[1] = a1 * inv; zc[2] = a2 * inv; zc[3] = a3 * inv;
  }

  // u_before: normalize own features; lane owns elements 4*lane..4*lane+3
  {
    v4f ux = *(const v4f*)(x + (size_t)node * DIM + 4 * lane);
    float ss = dot4(ux, ux);
    ss += __shfl_xor(ss, 1);
    ss += __shfl_xor(ss, 2);          // sum over the 4 lanes of this capsule
    float inv = 1.0f / fmaxf(sqrtf(ss), 1e-12f);
    ux *= inv;
    *(v4f*)(sm->ubf + 4 * lane) = ux;
  }
  __syncthreads();

  // ---------------- routing iterations ----------------
  const float prm = paramp[0];
  const int iters = iterp[0];
  const int half = lane >> 4;         // 0 or 1
  const int col = lane & 15;

  for (int it = 0; it < iters; ++it) {
    // p[m,c] = sum_dd z[m,c,dd] * u[c,dd]   (lane = m)
    float p[KCAPS];
    if (it == 0) {
#pragma unroll
      for (int c = 0; c < KCAPS; ++c) p[c] = 0.0f;
    } else {
      const v4f* z4 = (const v4f*)zr;
      const v4f* u4 = (const v4f*)sm->ub;
#pragma unroll
      for (int c = 0; c < KCAPS; ++c) {
        float acc = 0.0f;
#pragma unroll
        for (int j = 0; j < 4; ++j) acc += dot4(z4[c * 4 + j], u4[c * 4 + j]);
        p[c] = acc;
      }
    }

    // softmax over neighbors (32 lanes) per capsule -> p1
    float p1[KCAPS];
#pragma unroll
    for (int c = 0; c < KCAPS; ++c) {
      float mx = p[c];
#pragma unroll
      for (int o = 16; o >= 1; o >>= 1) mx = fmaxf(mx, __shfl_xor(mx, o));
      float e = __expf(p[c] - mx);
      float s = e;
#pragma unroll
      for (int o = 16; o >= 1; o >>= 1) s += __shfl_xor(s, o);
      p1[c] = e / s;
    }

    // softmax over capsules (in-lane) -> p2
    float p2[KCAPS];
    float mc = p[0];
#pragma unroll
    for (int c = 1; c < KCAPS; ++c) mc = fmaxf(mc, p[c]);
    float s2 = 0.0f;
#pragma unroll
    for (int c = 0; c < KCAPS; ++c) { p2[c] = __expf(p[c] - mc); s2 += p2[c]; }
    float r2 = 1.0f / s2;
#pragma unroll
    for (int c = 0; c < KCAPS; ++c) p2[c] *= r2;

    // stage [p1 | p2] for WMMA B operand
    {
      float* pr = sm->pb + lane * 16;
#pragma unroll
      for (int c = 0; c < KCAPS; ++c) { pr[c] = p1[c]; pr[8 + c] = p2[c]; }
    }
    __syncthreads();

    // U12[128,16] = Z^T[128,32] x [p1|p2][32,16] via v_wmma_f32_16x16x4_f32.
    // A tile (t,s): M=feature 16t+col, K = m = 4s + {0,1}/{2,3} (half splits).
    v2f B[8];
#pragma unroll
    for (int s = 0; s < 8; ++s) {
      B[s][0] = sm->pb[(4 * s + 2 * half) * 16 + col];
      B[s][1] = sm->pb[(4 * s + 2 * half + 1) * 16 + col];
    }
    v8f acc[8] = {};
#pragma unroll
    for (int t = 0; t < 8; ++t) {
#pragma unroll
      for (int s = 0; s < 8; ++s) {
        v2f A;
        A[0] = sm->z[(4 * s + 2 * half) * DIM + 16 * t + col];
        A[1] = sm->z[(4 * s + 2 * half + 1) * DIM + 16 * t + col];
        acc[t] = __builtin_amdgcn_wmma_f32_16x16x4_f32(
            false, A, false, B[s], (short)0, acc[t], false, false);
      }
    }

    // extract block diagonal: tile t == capsule t; N==t -> u1, N==8+t -> u2.
    // D VGPR v: lanes0-15 hold (M=v, N=lane); lanes16-31 hold (M=v+8, N=lane-16).
    {
      const int tm = lane & 7;
      float* dst = (lane & 8) ? sm->ut2 : sm->ut1;
      const int hi = (lane >= 16) ? 8 : 0;
#pragma unroll
      for (int t = 0; t < 8; ++t) {
        if (tm == t) {
#pragma unroll
          for (int v = 0; v < 8; ++v) dst[t * DD + hi + v] = acc[t][v];
        }
      }
    }
    __syncthreads();

    // u = prm*u1 + (1-prm)*u2 + u_before ; renormalize except last iter
    {
      v4f u1 = *(v4f*)(sm->ut1 + 4 * lane);
      v4f u2 = *(v4f*)(sm->ut2 + 4 * lane);
      v4f ub0 = *(v4f*)(sm->ubf + 4 * lane);
      v4f un = prm * u1 + (1.0f - prm) * u2 + ub0;
      if (it < iters - 1) {
        float ss = dot4(un, un);
        ss += __shfl_xor(ss, 1);
        ss += __shfl_xor(ss, 2);
        float inv = 1.0f / fmaxf(sqrtf(ss), 1e-12f);
        un *= inv;
      }
      *(v4f*)(sm->ub + 4 * lane) = un;
    }
    __syncthreads();
  }

  if (valid) {
    v4f un = *(v4f*)(sm->ub + 4 * lane);
    *(v4f*)(out + (size_t)node * DIM + 4 * lane) = un;
  }
}

extern "C" void kernel_launch(void* const* d_in, const int* in_sizes, int n_in,
                              void* d_out, int out_size, void* d_ws, size_t ws_size,
                              hipStream_t stream) {
  const float* x = (const float*)d_in[0];
  const int* nbr = (const int*)d_in[1];
  const float* prm = (const float*)d_in[2];
  const int* mit = (const int*)d_in[3];
  float* out = (float*)d_out;
  const int n = in_sizes[0] / DIM;
  const int blocks = (n + WPB - 1) / WPB;
  routing_caps_kernel<<<blocks, WPB * 32, 0, stream>>>(x, nbr, prm, mit, out, n);
}